// DiagonalTransfer_69707319214697
// MI455X (gfx1250) — compile-verified
//
#include <hip/hip_runtime.h>
#include <hip/hip_bf16.h>
#include <float.h>

// -------- CDNA5 async global->LDS path (guarded so compile never breaks) ----
#if defined(__gfx1250__) && __has_builtin(__builtin_amdgcn_global_load_async_to_lds_b32)
#define USE_ASYNC_LDS 1
typedef int v4i_t __attribute__((__vector_size__(4 * sizeof(int))));
typedef __attribute__((address_space(1))) int*    async_gptr;
typedef __attribute__((address_space(3))) int*    async_lptr;
typedef __attribute__((address_space(1))) v4i_t*  async_gptr128;
typedef __attribute__((address_space(3))) v4i_t*  async_lptr128;
#else
#define USE_ASYNC_LDS 0
#endif

#define P1_BLOCK 256
#define P2_BLOCK 256
#define VEC      4
#define P2_TILE  (P2_BLOCK * VEC)   // 1024 columns per pass-2 block
#define P2_ROWS  4

// ---------------------------------------------------------------------------
// Pass 1: per-column (k) logsumexp partials over a slice of rows.
// xx is row-major [N][K]; each thread owns 4 consecutive columns -> b128
// loads, 512B per wave per row. Sweep 1: max; sweep 2: scaled sum (re-read
// hits L0/L2; whole xx (32MB) is L2-resident on a 192MB L2).
// ---------------------------------------------------------------------------
__global__ __launch_bounds__(P1_BLOCK) void lse_pass1(
    const float* __restrict__ xx, float* __restrict__ mpart,
    float* __restrict__ spart, int N, int K, int rsl) {
  const int k    = (blockIdx.x * P1_BLOCK + threadIdx.x) * VEC;
  const int r    = blockIdx.y;
  const int rows = N / rsl;
  const float* __restrict__ p = xx + (size_t)r * rows * K + k;

  float4 m = make_float4(-FLT_MAX, -FLT_MAX, -FLT_MAX, -FLT_MAX);
#pragma unroll 4
  for (int j = 0; j < rows; ++j) {
    const float4 v = *(const float4*)(p + (size_t)j * K);
    m.x = fmaxf(m.x, v.x); m.y = fmaxf(m.y, v.y);
    m.z = fmaxf(m.z, v.z); m.w = fmaxf(m.w, v.w);
  }
  float4 s = make_float4(0.f, 0.f, 0.f, 0.f);
#pragma unroll 4
  for (int j = 0; j < rows; ++j) {
    const float4 v = *(const float4*)(p + (size_t)j * K);
    s.x += __expf(v.x - m.x); s.y += __expf(v.y - m.y);
    s.z += __expf(v.z - m.z); s.w += __expf(v.w - m.w);
  }
  *(float4*)(mpart + (size_t)r * K + k) = m;
  *(float4*)(spart + (size_t)r * K + k) = s;
}

// ---------------------------------------------------------------------------
// Combine the rsl partials per column; also precompute e^{diag[i]} - 1.
// ---------------------------------------------------------------------------
__global__ __launch_bounds__(256) void lse_combine(
    const float* __restrict__ mpart, const float* __restrict__ spart,
    const float* __restrict__ diag, float* __restrict__ mfull,
    float* __restrict__ sfull, float* __restrict__ expd1,
    int N, int K, int rsl) {
  const int idx = blockIdx.x * 256 + threadIdx.x;
  if (idx < K) {
    float m = -FLT_MAX;
    for (int r = 0; r < rsl; ++r)
      m = fmaxf(m, mpart[(size_t)r * K + idx]);
    float s = 0.0f;
    for (int r = 0; r < rsl; ++r)
      s += spart[(size_t)r * K + idx] * __expf(mpart[(size_t)r * K + idx] - m);
    mfull[idx] = m;
    sfull[idx] = s;
  }
  if (idx < N) expd1[idx] = __expf(diag[idx]) - 1.0f;
}

// ---------------------------------------------------------------------------
// out[i,k] = mm + log( s[k]*e^{m[k]-mm} + e^{x-mm} * (e^{d_i}-1) ),
//   mm = max(m[k], x + max(d_i,0))  -> all exp arguments <= 0 (stable; the
//   d_i<0 subtraction is safe: the remaining sum over N-1 states dominates).
// ---------------------------------------------------------------------------
__device__ __forceinline__ float lse_elem(float x, float m, float s,
                                          float dpos, float E) {
  const float mm = fmaxf(m, x + dpos);
  const float v  = s * __expf(m - mm) + __expf(x - mm) * E;
  return mm + __logf(v);
}

// Pass 2: 4 cols x 4 rows per thread; (m,s) tiles staged into LDS with CDNA5
// async global->LDS copies; xx/out moved as b128.
__global__ __launch_bounds__(P2_BLOCK) void lse_pass2(
    const float* __restrict__ xx, const float* __restrict__ diag,
    const float* __restrict__ mfull, const float* __restrict__ sfull,
    const float* __restrict__ expd1, float* __restrict__ out, int N, int K) {
  __shared__ float msh[P2_TILE];
  __shared__ float ssh[P2_TILE];
  const int t  = threadIdx.x;
  const int k  = blockIdx.x * P2_TILE + t * VEC;
  const int i0 = blockIdx.y * P2_ROWS;

#if USE_ASYNC_LDS
#if __has_builtin(__builtin_amdgcn_global_load_async_to_lds_b128)
  __builtin_amdgcn_global_load_async_to_lds_b128(
      (async_gptr128)(mfull + k), (async_lptr128)&msh[t * VEC], 0, 0);
  __builtin_amdgcn_global_load_async_to_lds_b128(
      (async_gptr128)(sfull + k), (async_lptr128)&ssh[t * VEC], 0, 0);
#else
#pragma unroll
  for (int c = 0; c < VEC; ++c) {
    __builtin_amdgcn_global_load_async_to_lds_b32(
        (async_gptr)(mfull + k + c), (async_lptr)&msh[t * VEC + c], 0, 0);
    __builtin_amdgcn_global_load_async_to_lds_b32(
        (async_gptr)(sfull + k + c), (async_lptr)&ssh[t * VEC + c], 0, 0);
  }
#endif
#if __has_builtin(__builtin_amdgcn_s_wait_asynccnt)
  __builtin_amdgcn_s_wait_asynccnt(0);
#else
  asm volatile("s_wait_asynccnt 0" ::: "memory");
#endif
#else
  *(float4*)&msh[t * VEC] = *(const float4*)(mfull + k);
  *(float4*)&ssh[t * VEC] = *(const float4*)(sfull + k);
#endif
  __syncthreads();

  const float4 m4 = *(const float4*)&msh[t * VEC];
  const float4 s4 = *(const float4*)&ssh[t * VEC];

#pragma unroll
  for (int r = 0; r < P2_ROWS; ++r) {
    const int   i    = i0 + r;
    const float d    = diag[i];
    const float dpos = fmaxf(d, 0.0f);
    const float E    = expd1[i];
    const float4 x   = *(const float4*)(xx + (size_t)i * K + k);
    float4 o;
    o.x = lse_elem(x.x, m4.x, s4.x, dpos, E);
    o.y = lse_elem(x.y, m4.y, s4.y, dpos, E);
    o.z = lse_elem(x.z, m4.z, s4.z, dpos, E);
    o.w = lse_elem(x.w, m4.w, s4.w, dpos, E);
    *(float4*)(out + (size_t)i * K + k) = o;
  }
}

// ---------------------------------------------------------------------------
extern "C" void kernel_launch(void* const* d_in, const int* in_sizes, int n_in,
                              void* d_out, int out_size, void* d_ws, size_t ws_size,
                              hipStream_t stream) {
  const float* diag = (const float*)d_in[0];
  const float* xx   = (const float*)d_in[1];
  float*       out  = (float*)d_out;

  const int N = in_sizes[0];           // 1024
  const int K = in_sizes[1] / N;       // 8192

  // Row-slice count: as large as the workspace allows (deterministic).
  int rsl = 16;
  while (rsl > 1 &&
         (size_t)(2u * rsl * K + 2u * K + N) * sizeof(float) > ws_size)
    rsl >>= 1;

  // Workspace layout (floats): mpart[rsl*K] | spart[rsl*K] | m[K] | s[K] | expd1[N]
  float* ws    = (float*)d_ws;
  float* mpart = ws;
  float* spart = mpart + (size_t)rsl * K;
  float* mfull = spart + (size_t)rsl * K;
  float* sfull = mfull + K;
  float* expd1 = sfull + K;

  lse_pass1<<<dim3(K / (P1_BLOCK * VEC), rsl), P1_BLOCK, 0, stream>>>(
      xx, mpart, spart, N, K, rsl);

  const int cmb = (((K > N) ? K : N) + 255) / 256;
  lse_combine<<<cmb, 256, 0, stream>>>(mpart, spart, diag, mfull, sfull,
                                       expd1, N, K, rsl);

  lse_pass2<<<dim3(K / P2_TILE, N / P2_ROWS), P2_BLOCK, 0, stream>>>(
      xx, diag, mfull, sfull, expd1, out, N, K);
}